// EntityBert_73873437491564
// MI455X (gfx1250) — compile-verified
//
#include <hip/hip_runtime.h>
#include <hip/hip_bf16.h>

// Problem dims
#define B_  256
#define L_  512
#define H_  768
#define C_  9
#define MTOT (B_*L_)    // 131072 rows

typedef __attribute__((ext_vector_type(2))) float v2f;
typedef __attribute__((ext_vector_type(8))) float v8f;

#if defined(__has_builtin)
#if __has_builtin(__builtin_amdgcn_wmma_f32_16x16x4_f32)
#define HAVE_WMMA_F32X4 1
#endif
#if __has_builtin(__builtin_amdgcn_s_wait_asynccnt)
#define WAIT_ASYNC0() __builtin_amdgcn_s_wait_asynccnt(0)
#endif
#endif
#ifndef WAIT_ASYNC0
#define WAIT_ASYNC0() asm volatile("s_wait_asynccnt 0" ::: "memory")
#endif

#define ROWS_PER_WG 128     // 8 waves x 16-row tiles
#define KCHUNK      16
#define A_STRIDE    20      // words: 16B-aligned rows, 16 distinct banks over r=0..15

// ASYNC DMA: global -> LDS, 16 bytes per lane. VDST operand carries the LDS
// byte offset (low 32 bits of the generic pointer == AS3 offset per aperture rules).
__device__ __forceinline__ void async_copy_b128(unsigned lds_off, const float* gaddr) {
    asm volatile("global_load_async_to_lds_b128 %0, %1, off"
                 :: "v"(lds_off), "v"(gaddr) : "memory");
}

// ---------------------------------------------------------------------------
// Kernel 1: emission GEMM  em[b,l,c] = feats[b,l,:] . W[:,c] + bias[c]
// Double-buffered async global->LDS A tiles + V_WMMA_F32_16X16X4_F32.
// ---------------------------------------------------------------------------
__global__ __launch_bounds__(256)
void emis_gemm_kernel(const float* __restrict__ feats,
                      const float* __restrict__ W,
                      const float* __restrict__ bias,
                      float* __restrict__ em)
{
    __shared__ __align__(16) float lA[2][ROWS_PER_WG * A_STRIDE]; // 2 x 10240 B
    __shared__ float lB9[H_ * C_];                                // 768x9 = 27648 B

    const int tid  = threadIdx.x;
    const int wave = tid >> 5;
    const int lane = tid & 31;
    const int n    = lane & 15;      // N index (and M index for A reads)
    const int kh   = lane >> 4;      // lane half: K+0/1 vs K+2/3
    const int wg_row = blockIdx.x * ROWS_PER_WG;

    // B column with same-address broadcast for padded lanes (n>=9 -> dup of n-9)
    const int  nb   = (n < C_) ? n : n - C_;
    const bool bval = (n < C_);

    // Stage compact W (768x9) into LDS once.
    for (int i = tid; i < H_ * C_; i += 256) lB9[i] = W[i];

    // Per-thread async-copy coordinates: 2 x b128 per chunk covers 128 rows x 16 K.
    const int r0 = tid >> 2;             // 0..63
    const int q  = (tid & 3) * 4;        // col quad
    const float* g0 = feats + (size_t)(wg_row + r0) * H_ + q;
    const float* g1 = feats + (size_t)(wg_row + 64 + r0) * H_ + q;

#if HAVE_WMMA_F32X4
    v8f acc = {};
#else
    float acc[8] = {0.f,0.f,0.f,0.f,0.f,0.f,0.f,0.f};
#endif

    // Prologue: DMA chunk 0 into buffer 0.
    {
        const unsigned d0 = (unsigned)(size_t)(const void*)&lA[0][r0 * A_STRIDE + q];
        const unsigned d1 = (unsigned)(size_t)(const void*)&lA[0][(64 + r0) * A_STRIDE + q];
        async_copy_b128(d0, g0);
        async_copy_b128(d1, g1);
    }

    int ibuf = 0;
    for (int kc = 0; kc < H_; kc += KCHUNK) {
        WAIT_ASYNC0();          // my DMA for current buffer landed in LDS
        __syncthreads();        // everyone's DMA landed (and W staging on iter 0)

        // Prefetch next chunk into the other buffer while we compute.
        if (kc + KCHUNK < H_) {
            const int nxt = ibuf ^ 1;
            const unsigned d0 = (unsigned)(size_t)(const void*)&lA[nxt][r0 * A_STRIDE + q];
            const unsigned d1 = (unsigned)(size_t)(const void*)&lA[nxt][(64 + r0) * A_STRIDE + q];
            async_copy_b128(d0, g0 + kc + KCHUNK);
            async_copy_b128(d1, g1 + kc + KCHUNK);
        }

        const float* A = &lA[ibuf][0];
#pragma unroll
        for (int kk = 0; kk < KCHUNK; kk += 4) {
            const int kb  = kk + 2 * kh;            // this lane's K base
            const int row = (wave << 4) + n;        // A: M = lane&15
            const int gk  = kc + kb;
            const float b0r = lB9[gk * C_ + nb];
            const float b1r = lB9[(gk + 1) * C_ + nb];
#if HAVE_WMMA_F32X4
            v2f a, b;
            a.x = A[row * A_STRIDE + kb];
            a.y = A[row * A_STRIDE + kb + 1];
            b.x = bval ? b0r : 0.0f;                // VGPR0: K=kb, VGPR1: K=kb+1
            b.y = bval ? b1r : 0.0f;
            acc = __builtin_amdgcn_wmma_f32_16x16x4_f32(
                      false, a, false, b, (short)0, acc, false, false);
#else
            const float bx = bval ? b0r : 0.0f;
            const float by = bval ? b1r : 0.0f;
#pragma unroll
            for (int v = 0; v < 8; ++v) {
                const int rr = (wave << 4) + v + 8 * kh;
                acc[v] += A[rr * A_STRIDE + kb] * bx + A[rr * A_STRIDE + kb + 1] * by;
            }
#endif
        }
        __syncthreads();        // all waves done reading ibuf before it is re-filled
        ibuf ^= 1;
    }

    // C/D layout: VGPR v holds M = v + 8*(lane>>4), N = lane&15
    const float bn = bval ? bias[n] : 0.0f;
#pragma unroll
    for (int v = 0; v < 8; ++v) {
        const int gr = wg_row + (wave << 4) + v + 8 * kh;
        if (bval) em[(size_t)gr * C_ + n] = acc[v] + bn;
    }
}

// ---------------------------------------------------------------------------
// Kernel 2: CRF forward logsumexp + Viterbi + numerator + backtrace.
// One thread per batch element; 9x9 transitions fully unrolled in registers.
// ---------------------------------------------------------------------------
__global__ __launch_bounds__(32)
void crf_scan_kernel(const float* __restrict__ em,
                     const float* __restrict__ start_t,
                     const float* __restrict__ end_t,
                     const float* __restrict__ trans,
                     const int*   __restrict__ tags,
                     const int*   __restrict__ lengths,
                     unsigned char* __restrict__ bps,   // (L-1, B, 16) bytes
                     float* __restrict__ llh_terms,     // (B,)
                     float* __restrict__ decoded)       // (B, L) as float
{
    const int b = blockIdx.x * blockDim.x + threadIdx.x;
    if (b >= B_) return;

    float tr[C_ * C_];
#pragma unroll
    for (int i = 0; i < C_ * C_; ++i) tr[i] = trans[i];
    float st[C_], en[C_];
#pragma unroll
    for (int c = 0; c < C_; ++c) { st[c] = start_t[c]; en[c] = end_t[c]; }

    const int len = lengths[b];
    const float* emb = em + (size_t)b * L_ * C_;

    float sc[C_], vs[C_];
#pragma unroll
    for (int c = 0; c < C_; ++c) {
        const float v = st[c] + emb[c];
        sc[c] = v; vs[c] = v;
    }

    int tag_prev = tags[b * L_ + 0];
    float numer = start_t[tag_prev] + emb[tag_prev];

    for (int t = 1; t < len; ++t) {
        float e[C_];
#pragma unroll
        for (int c = 0; c < C_; ++c) e[c] = emb[t * C_ + c];

        const int tg = tags[b * L_ + t];
        numer += trans[tag_prev * C_ + tg] + emb[t * C_ + tg];
        tag_prev = tg;

        float ns[C_], nv[C_];
#pragma unroll
        for (int j = 0; j < C_; ++j) {
            // forward logsumexp over previous states
            float m = -3.4e38f;
#pragma unroll
            for (int i = 0; i < C_; ++i) m = fmaxf(m, sc[i] + tr[i * C_ + j]);
            float s = 0.0f;
#pragma unroll
            for (int i = 0; i < C_; ++i) s += __expf(sc[i] + tr[i * C_ + j] - m);
            ns[j] = m + __logf(s) + e[j];

            // viterbi max/argmax (first-max semantics -> strict >)
            float vm = -3.4e38f; int arg = 0;
#pragma unroll
            for (int i = 0; i < C_; ++i) {
                const float v = vs[i] + tr[i * C_ + j];
                if (v > vm) { vm = v; arg = i; }
            }
            nv[j] = vm + e[j];
            bps[((size_t)(t - 1) * B_ + b) * 16 + j] = (unsigned char)arg;
        }
#pragma unroll
        for (int j = 0; j < C_; ++j) { sc[j] = ns[j]; vs[j] = nv[j]; }
    }

    numer += end_t[tag_prev];

    // logZ
    float m = -3.4e38f;
#pragma unroll
    for (int c = 0; c < C_; ++c) m = fmaxf(m, sc[c] + en[c]);
    float s = 0.0f;
#pragma unroll
    for (int c = 0; c < C_; ++c) s += __expf(sc[c] + en[c] - m);
    llh_terms[b] = numer - (m + __logf(s));

    // Viterbi terminal state
    float bm = -3.4e38f; int cur = 0;
#pragma unroll
    for (int c = 0; c < C_; ++c) {
        const float v = vs[c] + en[c];
        if (v > bm) { bm = v; cur = c; }
    }

    float* db = decoded + (size_t)b * L_;
    for (int t = len; t < L_; ++t) db[t] = 0.0f;   // masked positions -> 0
    for (int i = len - 2; i >= 0; --i) {
        db[i + 1] = (float)cur;
        cur = bps[((size_t)i * B_ + b) * 16 + cur];
    }
    db[0] = (float)cur;
}

// ---------------------------------------------------------------------------
// Kernel 3: mean of per-batch llh terms -> d_out[0]
// ---------------------------------------------------------------------------
__global__ __launch_bounds__(256)
void llh_reduce_kernel(const float* __restrict__ terms, float* __restrict__ out)
{
    __shared__ float s[256];
    const int t = threadIdx.x;
    s[t] = terms[t];
    __syncthreads();
    for (int o = 128; o > 0; o >>= 1) {
        if (t < o) s[t] += s[t + o];
        __syncthreads();
    }
    if (t == 0) out[0] = s[0] / (float)B_;
}

// ---------------------------------------------------------------------------
extern "C" void kernel_launch(void* const* d_in, const int* in_sizes, int n_in,
                              void* d_out, int out_size, void* d_ws, size_t ws_size,
                              hipStream_t stream) {
    const float* feats   = (const float*)d_in[0];
    const float* W_ff    = (const float*)d_in[1];
    const float* b_ff    = (const float*)d_in[2];
    const float* start_t = (const float*)d_in[3];
    const float* end_t   = (const float*)d_in[4];
    const float* trans   = (const float*)d_in[5];
    const int*   tags    = (const int*)d_in[6];
    const int*   lengths = (const int*)d_in[7];

    float* out = (float*)d_out;          // [0] = llh, [1..] = decoded (B*L)

    char* ws = (char*)d_ws;
    const size_t EM_BYTES  = (size_t)MTOT * C_ * sizeof(float);          // 4,718,592
    const size_t BPS_BYTES = (size_t)(L_ - 1) * B_ * 16;                 // 2,093,056
    float*         em    = (float*)ws;
    unsigned char* bps   = (unsigned char*)(ws + EM_BYTES);
    float*         terms = (float*)(ws + EM_BYTES + BPS_BYTES);

    emis_gemm_kernel<<<MTOT / ROWS_PER_WG, 256, 0, stream>>>(feats, W_ff, b_ff, em);
    crf_scan_kernel<<<B_ / 32, 32, 0, stream>>>(em, start_t, end_t, trans, tags,
                                                lengths, bps, terms, out + 1);
    llh_reduce_kernel<<<1, 256, 0, stream>>>(terms, out);
}